// AdditiveSynth_81003083203324
// MI455X (gfx1250) — compile-verified
//
#include <hip/hip_runtime.h>
#include <hip/hip_bf16.h>

typedef __attribute__((ext_vector_type(2))) float v2f;
typedef __attribute__((ext_vector_type(8))) float v8f;

#define NUM_B   2
#define NUM_V   8
#define NUM_H   64
#define NUM_F   500
#define NV      (NUM_B * NUM_V)        // 16
#define CH      (NV * NUM_H)           // 1024 channels (bv*64 + h)
#define NSAMP   (NUM_F * 64)           // 32000 per batch
#define NYQ     8000.0f
#define KOM     (6.28318530717958647692f / 16000.0f)  // 2*pi / SR
#define TWOPI_F 6.28318530717958647692f
#define INV2PI  0.15915494309189533577f
#define GAIN    0.02f

// Branchless mod 2*pi (inputs bounded ~<4e3 rad -> plenty of f32 headroom).
__device__ __forceinline__ float mod2pi(float x) {
  return x - TWOPI_F * floorf(x * INV2PI);
}

// Branchless sin: reduce to [0,2*pi) with one floor, then hardware v_sin_f32.
// Avoids libm's divergent Payne-Hanek slow path entirely.
__device__ __forceinline__ float fast_sin(float x) {
  float t = x * INV2PI;
  t -= floorf(t);
  return __sinf(t * TWOPI_F);
}

// ---------------------------------------------------------------------------
// K1: per (bv, frame): harmonic freqs, nyquist mask, normalize, amp scale.
// One wave per (bv,frame) pair, 2 harmonics per lane, wave32 shuffle reduce.
// Writes frame-major hf[frame][1024], ha[frame][1024] (ha has 501 frames,
// last duplicated for the overlap-add upsampler).
// ---------------------------------------------------------------------------
__global__ void k1_prep(const float* __restrict__ freq,
                        const float* __restrict__ harm,
                        const float* __restrict__ amp,
                        float* __restrict__ hf, float* __restrict__ ha) {
  int gtid = blockIdx.x * blockDim.x + threadIdx.x;
  int pair = gtid >> 5;                 // one wave per (bv, frame)
  if (pair >= NV * NUM_F) return;
  int lane = threadIdx.x & 31;
  int bv = pair / NUM_F;
  int f  = pair - bv * NUM_F;

  float fr = freq[bv * NUM_F + f];
  float am = amp [bv * NUM_F + f];

  float hfv[2], hw[2];
  float s = 0.0f;
#pragma unroll
  for (int u = 0; u < 2; ++u) {
    int h = lane + u * 32;
    float hfreq = fr * (float)(h + 1);                 // ratios = 1..64 exact
    float raw = harm[((bv * NUM_H + h) * NUM_F) + f];  // (b,v,h,f) layout
    float msk = (hfreq > NYQ) ? 0.0f : raw;
    hfv[u] = hfreq;   // NOTE: freq envelope uses UNmasked harm_freqs
    hw[u]  = msk;
    s += msk;
  }
#pragma unroll
  for (int off = 16; off >= 1; off >>= 1) s += __shfl_xor(s, off, 32);
  float den = (s == 0.0f) ? 1e-7f : s;
  float inv = 1.0f / den;

#pragma unroll
  for (int u = 0; u < 2; ++u) {
    int h = lane + u * 32;
    int c = bv * NUM_H + h;
    float hav = am * (hw[u] * inv);
    hf[f * CH + c] = hfv[u];
    ha[f * CH + c] = hav;
    if (f == NUM_F - 1) ha[NUM_F * CH + c] = hav;      // pad frame 500
  }
}

// Delta-phase sequence to scan: T'[0] = 32k*hf[0] (the 32 clamped lead-in
// samples), T'[j] = 32k*(hf[j-1]+hf[j]) for 1..499 (sum of one linear
// 64-sample segment), 0 beyond.
__device__ __forceinline__ float seg_delta(const float* __restrict__ hf,
                                           int idx, int ch) {
  int ia = idx - 1; ia = ia < 0 ? 0 : (ia > NUM_F - 1 ? NUM_F - 1 : ia);
  int ib = idx;     ib = ib > NUM_F - 1 ? NUM_F - 1 : ib;
  float fa = (idx >= 1) ? hf[ia * CH + ch] : 0.0f;
  float fb = hf[ib * CH + ch];
  float v = 32.0f * KOM * (fa + fb);
  return (idx < NUM_F) ? v : 0.0f;
}

// ---------------------------------------------------------------------------
// K2: phase-offset prefix scan over 500 frames x 1024 channels using
// V_WMMA_F32_16X16X4_F32 with a lower-triangular ones A-matrix.
// D[m,n] = sum_{k<=m} T'[base+k][ch_n]; 4 chained WMMAs cover 16 frames,
// register carry (mod 2*pi) links the 32 blocks. One wave per 16 channels.
// Writes frame-major O[frame][1024] (512 padded frames), values mod 2*pi.
// ---------------------------------------------------------------------------
__global__ void k2_scan(const float* __restrict__ hf, float* __restrict__ O) {
  int wg   = blockIdx.x * (blockDim.x >> 5) + (threadIdx.x >> 5);
  int lane = threadIdx.x & 31;
  int hi   = lane >> 4;             // lane half selects K pair {0,1}/{2,3}
  int n    = lane & 15;             // column  = channel within group
  int m    = n;                     // row     = frame within 16-block
  int ch   = wg * 16 + n;
  int kA   = hi ? 2 : 0;

  float carry = 0.0f;
  for (int blk = 0; blk < 32; ++blk) {
    int base = blk * 16;
    v8f acc = {};
#pragma unroll
    for (int g = 0; g < 4; ++g) {
      int k0 = 4 * g + kA;
      v2f a, bb;
      a.x = (k0     <= m) ? 1.0f : 0.0f;     // triangular select, exact in f32
      a.y = (k0 + 1 <= m) ? 1.0f : 0.0f;
      bb.x = seg_delta(hf, base + k0,     ch);
      bb.y = seg_delta(hf, base + k0 + 1, ch);
      acc = __builtin_amdgcn_wmma_f32_16x16x4_f32(
          false, a, false, bb, (short)0, acc, false, false);
    }
#pragma unroll
    for (int r = 0; r < 8; ++r) acc[r] += carry;
    float tot = __shfl(acc[7], 16 + n, 32);  // D[15,n] lives in lane 16+n
#pragma unroll
    for (int r = 0; r < 8; ++r)
      O[(base + hi * 8 + r) * CH + ch] = mod2pi(acc[r]);
    carry = mod2pi(tot);
  }
}

// ---------------------------------------------------------------------------
// K3: per-sample synthesis + channel reduction. Each wave owns 16 aligned
// samples (rows of D). Loop over 128 groups of 4 channels: lanes compute
// sin(phase)*amp into the WMMA A operand, B = ones(4x16), C accumulates;
// after 128 WMMAs column 0 of D holds the 16 audio samples (v and h summed).
// Closed-form phase: O(j) + k*((r+1)*hfA + (hfB-hfA)*(r+1)^2/128).
// ---------------------------------------------------------------------------
__global__ void k3_synth(const float* __restrict__ hf,
                         const float* __restrict__ ha,
                         const float* __restrict__ O,
                         float* __restrict__ out) {
  int wid  = blockIdx.x * (blockDim.x >> 5) + (threadIdx.x >> 5);
  if (wid >= (NUM_B * NSAMP) / 16) return;
  int lane = threadIdx.x & 31;
  int hi   = lane >> 4;
  int m    = lane & 15;

  int sbase = wid * 16;
  int b  = sbase / NSAMP;
  int t0 = sbase - b * NSAMP;        // multiple of 16 -> region uniform/wave
  int i  = t0 + m;

  int jA, jB, jO, r; bool hasO;
  if (t0 < 32)               { jA = 0;   jB = 0;   jO = 0;   r = i;                hasO = false; }
  else if (t0 >= NSAMP - 32) { jA = 499; jB = 499; jO = 499; r = i - (NSAMP - 32); hasO = true;  }
  else { jA = (i - 32) >> 6; jB = jA + 1; jO = jA; r = (i - 32) & 63;             hasO = true;  }

  float rp1 = (float)(r + 1);
  float kr  = KOM * rp1;                                 // hoisted
  float kq  = KOM * rp1 * rp1 * (1.0f / 128.0f);         // hoisted
  float wl  = (float)(2 * r + 1) * (1.0f / 128.0f);      // per-sample lerp w

  int jb  = t0 >> 6;                                     // amp OLA frame
  int r64 = i & 63;
  float cw = __cosf(3.14159265358979323846f * (float)r64 * (1.0f / 64.0f));
  float w0 = 0.5f - 0.5f * cw;                           // win[r64]
  float w1 = 0.5f + 0.5f * cw;                           // win[r64+64]

  const float* pHfA = hf + jA * CH + b * 512;
  const float* pHfB = hf + jB * CH + b * 512;
  const float* pHaA = ha + jb * CH + b * 512;
  const float* pHaB = ha + (jb + 1) * CH + b * 512;
  const float* pO   = O  + jO * CH + b * 512;

  v2f ones; ones.x = 1.0f; ones.y = 1.0f;
  v8f acc = {};
  for (int ci = 0; ci < 128; ++ci) {
    int c0 = 4 * ci + (hi ? 2 : 0);
    v2f fa = *(const v2f*)(pHfA + c0);
    v2f fb = *(const v2f*)(pHfB + c0);
    v2f aa = *(const v2f*)(pHaA + c0);
    v2f ab = *(const v2f*)(pHaB + c0);
    v2f oj;
    if (hasO) oj = *(const v2f*)(pO + c0); else { oj.x = 0.0f; oj.y = 0.0f; }

    v2f a;
    {
      float df   = fb.x - fa.x;
      float fenv = fa.x + df * wl;
      float ampv = aa.x * w1 + ab.x * w0;
      float ph   = oj.x + kr * fa.x + kq * df;
      a.x = (fenv > NYQ) ? 0.0f : ampv * fast_sin(ph);
    }
    {
      float df   = fb.y - fa.y;
      float fenv = fa.y + df * wl;
      float ampv = aa.y * w1 + ab.y * w0;
      float ph   = oj.y + kr * fa.y + kq * df;
      a.y = (fenv > NYQ) ? 0.0f : ampv * fast_sin(ph);
    }
    acc = __builtin_amdgcn_wmma_f32_16x16x4_f32(
        false, a, false, ones, (short)0, acc, false, false);
  }

  if (m == 0) {                       // lanes 0 / 16 hold column N=0 (M 0-7 / 8-15)
    float* dst = out + b * NSAMP + t0 + hi * 8;
#pragma unroll
    for (int rr = 0; rr < 8; ++rr) dst[rr] = acc[rr] * GAIN;
  }
}

// ---------------------------------------------------------------------------
extern "C" void kernel_launch(void* const* d_in, const int* in_sizes, int n_in,
                              void* d_out, int out_size, void* d_ws, size_t ws_size,
                              hipStream_t stream) {
  (void)in_sizes; (void)n_in; (void)out_size; (void)ws_size;
  const float* freq = (const float*)d_in[0];   // (2,8,500)
  const float* harm = (const float*)d_in[1];   // (2,8,64,500)
  const float* amp  = (const float*)d_in[2];   // (2,8,500)
  float* out = (float*)d_out;                  // (2,32000)

  // workspace: hf[500][1024] | ha[501][1024] | O[512][1024]  (~6.2 MB)
  float* ws = (float*)d_ws;
  float* hf = ws;
  float* ha = hf + (size_t)NUM_F * CH;
  float* Ob = ha + (size_t)(NUM_F + 1) * CH;

  k1_prep<<<dim3(1000), dim3(256), 0, stream>>>(freq, harm, amp, hf, ha);
  k2_scan<<<dim3(16),   dim3(128), 0, stream>>>(hf, Ob);
  k3_synth<<<dim3(1000), dim3(128), 0, stream>>>(hf, ha, Ob, out);
}